// Policy_Network_38774964748846
// MI455X (gfx1250) — compile-verified
//
#include <hip/hip_runtime.h>
#include <hip/hip_bf16.h>
#include <stdint.h>

// dims (fixed by the reference)
#define B_  128
#define S_  32
#define DW  300
#define DR  256
#define DE  256
#define H_  512
#define R_  512
#define E_  50000
#define A_  256
#define ACT 512          // DR + DE
#define INP 768          // H + DR
#define NEGV (-1e9f)

// LDS strides (pad 4 floats -> bank shift of 4 per row: conflict-free for the
// 16x4 A-fragment ds_load_b64 pattern, and 16B-aligned rows for async B128)
#define SA_STRIDE  260   // raq stage: 32 x 260 floats
#define SX_STRIDE  516   // X1 stage:  32 x 516 floats

typedef float v2f __attribute__((ext_vector_type(2)));
typedef float v8f __attribute__((ext_vector_type(8)));

// ---------------------------------------------------------------------------
// Kernel 1: saq = tanh(transformer_output @ W_step + b_step)   [B*S, DR]
// ---------------------------------------------------------------------------
__global__ __launch_bounds__(256) void k_saq(const float* __restrict__ TO,
                                             const float* __restrict__ Wstep,
                                             const float* __restrict__ bstep,
                                             float* __restrict__ saq) {
  __shared__ float sTO[DW];
  const int m = blockIdx.x;
  const int t = threadIdx.x;
  for (int i = t; i < DW; i += 256) sTO[i] = TO[(size_t)m * DW + i];
  __syncthreads();
  float acc = bstep[t];
#pragma unroll 4
  for (int k = 0; k < DW; ++k) acc += sTO[k] * Wstep[(size_t)k * DR + t];
  saq[(size_t)m * DR + t] = tanhf(acc);
}

// ---------------------------------------------------------------------------
// Kernel 2: PH1[b,n] = path_hidden[b,:] @ W1[0:H,:] + b1[n]    [B, ACT]
// (path part of layer-1 input is identical across all r of a batch element)
// ---------------------------------------------------------------------------
__global__ __launch_bounds__(256) void k_ph1(const float* __restrict__ ph,
                                             const float* __restrict__ W1,
                                             const float* __restrict__ b1,
                                             float* __restrict__ PH1) {
  const int b = blockIdx.x;
  const int t = threadIdx.x;
  for (int n = t; n < ACT; n += 256) {
    float acc = b1[n];
#pragma unroll 4
    for (int k = 0; k < H_; ++k) acc += ph[(size_t)b * H_ + k] * W1[(size_t)k * ACT + n];
    PH1[(size_t)b * ACT + n] = acc;
  }
}

// ---------------------------------------------------------------------------
// Kernel 3: fused relation-aware attention.
// scores[b,r,s] = dot(saq[b,s,:]*w_att, rel[r,:]) + b_att ; mask ; softmax(s)
// raq[b,r,:]    = sum_s alpha[s] * saq[b,s,:]
// grid (B, R/8), 256 threads.  saq[b] staged in LDS, wave0 shuffle softmax.
// ---------------------------------------------------------------------------
__global__ __launch_bounds__(256) void k_attn(const float* __restrict__ saq,
                                              const float* __restrict__ watt,
                                              const float* __restrict__ batt,
                                              const float* __restrict__ rel,
                                              const uint8_t* __restrict__ qmask,
                                              float* __restrict__ raq) {
  __shared__ float s_saq[S_ * 257];
  __shared__ float s_prod[DR];
  __shared__ float s_red[256];
  __shared__ float s_alpha[S_];
  const int b = blockIdx.x;
  const int r0 = blockIdx.y * 8;
  const int t = threadIdx.x;
  for (int i = t; i < S_ * DR; i += 256)
    s_saq[(i >> 8) * 257 + (i & 255)] = saq[((size_t)b * S_ + (i >> 8)) * DR + (i & 255)];
  const float battv = batt[0];
  __syncthreads();

  const int s = t >> 3, part = t & 7;
  for (int ri = 0; ri < 8; ++ri) {
    const int r = r0 + ri;
    s_prod[t] = watt[t] * rel[(size_t)r * DR + t];
    __syncthreads();
    float p = 0.f;
#pragma unroll
    for (int dd = 0; dd < 32; ++dd) {
      const int d = part * 32 + dd;
      p += s_saq[s * 257 + d] * s_prod[d];
    }
    s_red[t] = p;
    __syncthreads();
    if (t < S_) {                       // wave 0: one lane per sequence position
      float sc = battv;
#pragma unroll
      for (int j = 0; j < 8; ++j) sc += s_red[t * 8 + j];
      if (qmask[b * S_ + t]) sc = NEGV;
      float mx = sc;
#pragma unroll
      for (int off = 16; off > 0; off >>= 1) mx = fmaxf(mx, __shfl_xor(mx, off, 32));
      const float e = __expf(sc - mx);
      float sum = e;
#pragma unroll
      for (int off = 16; off > 0; off >>= 1) sum += __shfl_xor(sum, off, 32);
      s_alpha[t] = e / sum;
    }
    __syncthreads();
    float acc = 0.f;
#pragma unroll
    for (int ss = 0; ss < S_; ++ss) acc += s_alpha[ss] * s_saq[ss * 257 + t];
    raq[((size_t)(b * R_ + r)) * DR + t] = acc;
    __syncthreads();
  }
}

// ---------------------------------------------------------------------------
// Kernel 4: fused policy MLP using V_WMMA_F32_16X16X4_F32.
//   X1 = relu(PH1[b] + raq @ W1[H:, :])    (K = 256)
//   X2 = X1 @ W2 + b2                      (K = 512), X1 kept in LDS
// raq rows staged memory->LDS with GLOBAL_LOAD_ASYNC_TO_LDS_B128 (ASYNCcnt).
// ---------------------------------------------------------------------------
__global__ __launch_bounds__(256) void k_mlp(const float* __restrict__ raq,
                                             const float* __restrict__ PH1,
                                             const float* __restrict__ W1,
                                             const float* __restrict__ W2,
                                             const float* __restrict__ b2,
                                             float* __restrict__ X2) {
  __shared__ float sA[32 * SA_STRIDE];    // raq rows (K=256)
  __shared__ float sX1[32 * SX_STRIDE];   // layer-1 activations (512)
  const int m0 = blockIdx.x * 32;
  const int b = m0 >> 9;            // 512 rows per batch element, 32 | 512
  const int t = threadIdx.x;
  const int wave = t >> 5, lane = t & 31;
  const int rsel = lane & 15;           // row (A) / col (B,C) selector
  const int hi = lane >> 4;             // lanes 16-31: K+2/K+3 (A,B), M+8 (C/D)
  const int khalf = hi * 2;
  const int colb = wave * 64;

  // ---- async DMA: raq rows -> LDS (2048 x 16B chunks, 8 per thread) ----
  for (int c = t; c < 32 * (DR / 4); c += 256) {
    const int row = c >> 6;                       // 64 chunks per row
    const int ch = (c & 63) * 4;                  // float offset in row
    const uint32_t lds = (uint32_t)(uintptr_t)(&sA[row * SA_STRIDE + ch]);
    const uint64_t gp =
        (uint64_t)(uintptr_t)(raq + (size_t)(m0 + row) * DR + ch);
    asm volatile("global_load_async_to_lds_b128 %0, %1, off"
                 :: "v"(lds), "v"(gp) : "memory");
  }
  asm volatile("s_wait_asynccnt 0" ::: "memory");
  __syncthreads();

  // ---- layer 1: K = 256 over raq part of W1 (rows H..H+255) ----
  v8f acc[2][4] = {};
  for (int k0 = 0; k0 < DR; k0 += 4) {
    v2f a[2];
#pragma unroll
    for (int rt = 0; rt < 2; ++rt) {
      const int row = rt * 16 + rsel;
      a[rt].x = sA[row * SA_STRIDE + k0 + khalf];
      a[rt].y = sA[row * SA_STRIDE + k0 + khalf + 1];
    }
    const float* w = W1 + (size_t)(H_ + k0 + khalf) * ACT;
    __builtin_prefetch(w + (size_t)4 * ACT, 0, 3);   // near-scope prefetch
    v2f bf[4];
#pragma unroll
    for (int ct = 0; ct < 4; ++ct) {
      const int col = colb + ct * 16 + rsel;
      bf[ct].x = w[col];
      bf[ct].y = w[ACT + col];
    }
#pragma unroll
    for (int rt = 0; rt < 2; ++rt)
#pragma unroll
      for (int ct = 0; ct < 4; ++ct)
        acc[rt][ct] = __builtin_amdgcn_wmma_f32_16x16x4_f32(
            false, a[rt], false, bf[ct], (short)0, acc[rt][ct], false, false);
  }
  // epilogue 1: add shared path contribution (bias folded in), ReLU -> LDS
#pragma unroll
  for (int rt = 0; rt < 2; ++rt)
#pragma unroll
    for (int ct = 0; ct < 4; ++ct) {
      const int col = colb + ct * 16 + rsel;
      const float ph1v = PH1[(size_t)b * ACT + col];
#pragma unroll
      for (int j = 0; j < 8; ++j) {
        const int row = rt * 16 + j + hi * 8;
        const float v = acc[rt][ct][j] + ph1v;
        sX1[row * SX_STRIDE + col] = v > 0.f ? v : 0.f;
      }
    }
  __syncthreads();

  // ---- layer 2: K = 512 from LDS ----
  v8f acc2[2][4] = {};
  for (int k0 = 0; k0 < ACT; k0 += 4) {
    v2f a[2];
#pragma unroll
    for (int rt = 0; rt < 2; ++rt) {
      const int row = rt * 16 + rsel;
      a[rt].x = sX1[row * SX_STRIDE + k0 + khalf];
      a[rt].y = sX1[row * SX_STRIDE + k0 + khalf + 1];
    }
    const float* w = W2 + (size_t)(k0 + khalf) * ACT;
    __builtin_prefetch(w + (size_t)4 * ACT, 0, 3);
    v2f bf[4];
#pragma unroll
    for (int ct = 0; ct < 4; ++ct) {
      const int col = colb + ct * 16 + rsel;
      bf[ct].x = w[col];
      bf[ct].y = w[ACT + col];
    }
#pragma unroll
    for (int rt = 0; rt < 2; ++rt)
#pragma unroll
      for (int ct = 0; ct < 4; ++ct)
        acc2[rt][ct] = __builtin_amdgcn_wmma_f32_16x16x4_f32(
            false, a[rt], false, bf[ct], (short)0, acc2[rt][ct], false, false);
  }
  // epilogue 2: + b2, store X2
#pragma unroll
  for (int rt = 0; rt < 2; ++rt)
#pragma unroll
    for (int ct = 0; ct < 4; ++ct) {
      const int col = colb + ct * 16 + rsel;
      const float bias = b2[col];
#pragma unroll
      for (int j = 0; j < 8; ++j) {
        const int row = rt * 16 + j + hi * 8;
        X2[(size_t)(m0 + row) * ACT + col] = acc2[rt][ct][j] + bias;
      }
    }
}

// ---------------------------------------------------------------------------
// Kernel 5: logits[b,a] = dot(X2[b, r_space[b,a], :], [rel[r];ent[e]]),
// action-mask, softmax over A.  One block per b, thread = action.
// ---------------------------------------------------------------------------
__global__ __launch_bounds__(256) void k_logits(const float* __restrict__ X2,
                                                const float* __restrict__ rel,
                                                const float* __restrict__ ent,
                                                const int* __restrict__ rspace,
                                                const int* __restrict__ espace,
                                                const float* __restrict__ amask,
                                                float* __restrict__ out) {
  __shared__ float red[256];
  const int b = blockIdx.x;
  const int a = threadIdx.x;
  const int r = rspace[b * A_ + a];
  const int e = espace[b * A_ + a];
  const float* xr = X2 + ((size_t)(b * R_ + r)) * ACT;
  const float* rr = rel + (size_t)r * DR;
  const float* er = ent + (size_t)e * DE;
  float acc = 0.f;
#pragma unroll 4
  for (int d = 0; d < DR; ++d) acc += xr[d] * rr[d];
#pragma unroll 4
  for (int d = 0; d < DE; ++d) acc += xr[DR + d] * er[d];
  if (!(amask[b * A_ + a] > 0.f)) acc = NEGV;
  red[a] = acc;
  __syncthreads();
  for (int sft = 128; sft > 0; sft >>= 1) {
    if (a < sft) red[a] = fmaxf(red[a], red[a + sft]);
    __syncthreads();
  }
  const float mx = red[0];
  __syncthreads();
  const float ev = __expf(acc - mx);
  red[a] = ev;
  __syncthreads();
  for (int sft = 128; sft > 0; sft >>= 1) {
    if (a < sft) red[a] += red[a + sft];
    __syncthreads();
  }
  out[b * A_ + a] = ev / red[0];
}

// ---------------------------------------------------------------------------
extern "C" void kernel_launch(void* const* d_in, const int* in_sizes, int n_in,
                              void* d_out, int out_size, void* d_ws, size_t ws_size,
                              hipStream_t stream) {
  (void)in_sizes; (void)n_in; (void)out_size; (void)ws_size;
  const float*   TO    = (const float*)d_in[0];
  const uint8_t* qmask = (const uint8_t*)d_in[1];
  const float*   Wstep = (const float*)d_in[2];
  const float*   bstep = (const float*)d_in[3];
  const float*   watt  = (const float*)d_in[4];
  const float*   batt  = (const float*)d_in[5];
  const float*   rel   = (const float*)d_in[6];
  const float*   ent   = (const float*)d_in[7];
  const float*   ph    = (const float*)d_in[8];
  const float*   W1    = (const float*)d_in[9];
  const float*   b1    = (const float*)d_in[10];
  const float*   W2    = (const float*)d_in[11];
  const float*   b2    = (const float*)d_in[12];
  const int*     rsp   = (const int*)d_in[13];
  const int*     esp   = (const int*)d_in[14];
  const float*   amask = (const float*)d_in[15];
  float* out = (float*)d_out;

  float* ws  = (float*)d_ws;
  float* saq = ws;                                  // [4096, 256]   4 MB
  float* raq = saq + (size_t)B_ * S_ * DR;          // [65536, 256] 64 MB
  float* PH1 = raq + (size_t)B_ * R_ * DR;          // [128, 512]  256 KB
  float* X2  = PH1 + (size_t)B_ * ACT;              // [65536, 512]128 MB

  k_saq   <<<dim3(B_ * S_),      256, 0, stream>>>(TO, Wstep, bstep, saq);
  k_ph1   <<<dim3(B_),           256, 0, stream>>>(ph, W1, b1, PH1);
  k_attn  <<<dim3(B_, R_ / 8),   256, 0, stream>>>(saq, watt, batt, rel, qmask, raq);
  k_mlp   <<<dim3(B_ * R_ / 32), 256, 0, stream>>>(raq, PH1, W1, W2, b2, X2);
  k_logits<<<dim3(B_),           256, 0, stream>>>(X2, rel, ent, rsp, esp, amask, out);
}